// StudentQValueNet_35699768164384
// MI455X (gfx1250) — compile-verified
//
#include <hip/hip_runtime.h>
#include <math.h>
#include <stdint.h>

#define NN 50000
#define EE 800000
#define NG 16

typedef float v2f __attribute__((ext_vector_type(2)));
typedef float v8f __attribute__((ext_vector_type(8)));
typedef unsigned int u32x4 __attribute__((ext_vector_type(4)));
typedef int i32x4 __attribute__((ext_vector_type(4)));
typedef int i32x8 __attribute__((ext_vector_type(8)));

#define LDS_PITCH 130   // 128 dwords + TDM pad_amount=2 dwords per row (bank spread)

__device__ __forceinline__ float lrelu_f(float v) { return v > 0.f ? v : 0.2f * v; }

// ---------------------------------------------------------------------------
// TDM: DMA a rows x 128 f32 tile (row-major, row stride 128 floats in memory)
// into LDS at lds_off, hardware-padding +2 dwords after every 128-dword row.
// Issue from ONE wave; completion via s_wait_tensorcnt; publish via barrier.
__device__ __forceinline__ void tdm_load_tile(const float* gptr, unsigned int lds_off, int rows)
{
    unsigned long long ga = (unsigned long long)(uintptr_t)gptr;
    u32x4 g0;
    g0[0] = 1u;                                              // count=1 (valid D#)
    g0[1] = lds_off;                                         // lds_addr (bytes)
    g0[2] = (unsigned int)(ga & 0xffffffffu);                // global_addr[31:0]
    g0[3] = (unsigned int)((ga >> 32) & 0x01ffffffu)         // global_addr[56:32]
          | (2u << 30);                                      // type=2 ("image")
    i32x8 g1;
    g1[0] = (int)((2u << 16)     // data_size = 4B
          | (1u << 20)           // pad_enable
          | (6u << 22)           // pad_interval: 2^(6+1)=128 dwords
          | (1u << 25));         // pad_amount: 2 dwords
    g1[1] = (int)(128u << 16);                // tensor_dim0[15:0] = 128 cols
    g1[2] = (int)((unsigned)rows << 16);      // tensor_dim1[15:0] = rows
    g1[3] = (int)(128u << 16);                // tile_dim0 = 128
    g1[4] = rows;                             // tile_dim1 = rows (tile_dim2 = 0)
    g1[5] = 128;                              // tensor_dim0_stride = 128 elems
    g1[6] = 0;
    g1[7] = 0;
    i32x4 z4 = {0, 0, 0, 0};
#if __has_include(<hip/amd_detail/amd_gfx1250_TDM.h>)
    i32x8 z8 = {0, 0, 0, 0, 0, 0, 0, 0};
    __builtin_amdgcn_tensor_load_to_lds(g0, g1, z4, z4, z8, 0);
#else
    __builtin_amdgcn_tensor_load_to_lds(g0, g1, z4, z4, 0);
#endif
}

// ---------------------------------------------------------------------------
// C[r,0:128] = act( (A[r,:] (+ states[r]*a0)) @ W^T + bias ), W 128x128 row-major.
// One wave per 16-row strip; 8 f32 WMMA 16x16x4 accumulators; W staged via TDM
// into pad-striped LDS (pitch 130 dwords).
template <bool PREADD, bool ACT, bool ACCLOAD, bool HASBIAS>
__global__ __launch_bounds__(256) void gemm128_kernel(
    const float* A, const float* __restrict__ W,
    const float* __restrict__ bias, const float* __restrict__ states,
    const float* __restrict__ a0p, float* C, int nrows)
{
    extern __shared__ float lw[];   // 128 * 130 * 4 = 66560 B, lw[j*130 + k] = W[j][k]
    const int tid = threadIdx.x;
    if (tid < 32)                   // wave 0 drives the Tensor Data Mover
        tdm_load_tile(W, (unsigned int)(uintptr_t)lw, 128);

    const int lane = tid & 31;
    const int hi = lane >> 4;   // 0|1
    const int lr = lane & 15;   // 0..15
    const int m0 = blockIdx.x * 128 + (tid >> 5) * 16;

    int rowA = m0 + lr;
    if (rowA >= nrows) rowA = nrows - 1;     // clamp: keep EXEC all-ones for WMMA
    const float* arow = A + (size_t)rowA * 128;

    float sadd = 0.f;
    if (PREADD) sadd = states[rowA] * a0p[0];

    v8f acc[8];
    #pragma unroll
    for (int nt = 0; nt < 8; ++nt) {
        #pragma unroll
        for (int v = 0; v < 8; ++v) {
            if (ACCLOAD) {
                int r = m0 + v + 8 * hi;
                if (r >= nrows) r = nrows - 1;
                acc[nt][v] = C[(size_t)r * 128 + nt * 16 + lr];
            } else {
                acc[nt][v] = 0.f;
            }
        }
    }

    if (tid < 32) __builtin_amdgcn_s_wait_tensorcnt(0);
    __syncthreads();                // publish TDM-written LDS to all waves

    for (int kb = 0; kb < 128; kb += 4) {
        const int kk = kb + 2 * hi;                  // A frag: lane holds K = kk, kk+1
        v2f av = *(const v2f*)(arow + kk);
        if (PREADD) { av.x += sadd; av.y += sadd; }
        const float* b0 = lw + (size_t)lr * LDS_PITCH + kk;  // B[kk..kk+1, n0+lr] = W[n0+lr][kk..]
        #pragma unroll
        for (int nt = 0; nt < 8; ++nt) {
            v2f bv = *(const v2f*)(b0 + nt * 16 * LDS_PITCH);
            acc[nt] = __builtin_amdgcn_wmma_f32_16x16x4_f32(
                false, av, false, bv, (short)0, acc[nt], false, false);
        }
    }

    #pragma unroll
    for (int nt = 0; nt < 8; ++nt) {
        const int col = nt * 16 + lr;
        const float bb = HASBIAS ? bias[col] : 0.f;
        #pragma unroll
        for (int v = 0; v < 8; ++v) {
            const int r = m0 + v + 8 * hi;           // C layout: M = v + 8*hi
            if (r < nrows) {
                float val = acc[nt][v] + bb;
                if (ACT) val = lrelu_f(val);
                C[(size_t)r * 128 + col] = val;
            }
        }
    }
}

// ---------------------------------------------------------------------------
// hh[r,0:192] = lrelu( A[r,:] @ M0^T + w[batch[r], :] ), M0 192x128 staged via TDM.
__global__ __launch_bounds__(256) void gemm192_kernel(
    const float* __restrict__ A, const float* __restrict__ M0,
    const float* __restrict__ wb, const int* __restrict__ batch,
    float* __restrict__ C, int nrows)
{
    extern __shared__ float lw[];   // 192 * 130 * 4 = 99840 B
    const int tid = threadIdx.x;
    if (tid < 32)
        tdm_load_tile(M0, (unsigned int)(uintptr_t)lw, 192);

    const int lane = tid & 31;
    const int hi = lane >> 4, lr = lane & 15;
    const int m0 = blockIdx.x * 128 + (tid >> 5) * 16;

    int rowA = m0 + lr;
    if (rowA >= nrows) rowA = nrows - 1;
    const float* arow = A + (size_t)rowA * 128;

    v8f acc[12];
    #pragma unroll
    for (int nt = 0; nt < 12; ++nt)
        #pragma unroll
        for (int v = 0; v < 8; ++v) acc[nt][v] = 0.f;

    if (tid < 32) __builtin_amdgcn_s_wait_tensorcnt(0);
    __syncthreads();

    for (int kb = 0; kb < 128; kb += 4) {
        const int kk = kb + 2 * hi;
        v2f av = *(const v2f*)(arow + kk);
        const float* b0 = lw + (size_t)lr * LDS_PITCH + kk;
        #pragma unroll
        for (int nt = 0; nt < 12; ++nt) {
            v2f bv = *(const v2f*)(b0 + nt * 16 * LDS_PITCH);
            acc[nt] = __builtin_amdgcn_wmma_f32_16x16x4_f32(
                false, av, false, bv, (short)0, acc[nt], false, false);
        }
    }

    int bidx[8];
    #pragma unroll
    for (int v = 0; v < 8; ++v) {
        int r = m0 + v + 8 * hi;
        bidx[v] = (r < nrows) ? batch[r] : 0;
    }
    #pragma unroll
    for (int nt = 0; nt < 12; ++nt) {
        const int col = nt * 16 + lr;
        #pragma unroll
        for (int v = 0; v < 8; ++v) {
            const int r = m0 + v + 8 * hi;
            if (r < nrows)
                C[(size_t)r * 192 + col] = lrelu_f(acc[nt][v] + wb[bidx[v] * 192 + col]);
        }
    }
}

// ---------------------------------------------------------------------------
__global__ void split_kernel(const float* __restrict__ x, float* __restrict__ x1,
                             float* __restrict__ x2, int nrows)
{
    int t = blockIdx.x * blockDim.x + threadIdx.x;
    int i = t >> 5, p = t & 31;
    if (i >= nrows) return;
    float4 a = *(const float4*)&x[(size_t)i * 256 + p * 4];
    float4 b = *(const float4*)&x[(size_t)i * 256 + 128 + p * 4];
    *(float4*)&x1[(size_t)i * 128 + p * 4] = a;
    *(float4*)&x2[(size_t)i * 128 + p * 4] = b;
}

__global__ void zero4_kernel(float4* __restrict__ p, long n4)
{
    long t = (long)blockIdx.x * blockDim.x + threadIdx.x;
    if (t < n4) p[t] = make_float4(0.f, 0.f, 0.f, 0.f);
}

// agg[dst[e]] += ew[e] * h[src[e]]  (32 lanes/edge, float4 per lane, f32 atomics in L2)
__global__ void scatter_kernel(const float* __restrict__ h, const float* __restrict__ ew,
                               const int* __restrict__ src, const int* __restrict__ dst,
                               float* agg, int nedges)
{
    int t = blockIdx.x * blockDim.x + threadIdx.x;
    int e = t >> 5, part = t & 31;
    if (e >= nedges) return;
    int s = src[e], d = dst[e];
    float wgt = ew[e];
    float4 v = *(const float4*)&h[(size_t)s * 128 + part * 4];
    float* dp = &agg[(size_t)d * 128 + part * 4];
    unsafeAtomicAdd(dp + 0, wgt * v.x);
    unsafeAtomicAdd(dp + 1, wgt * v.y);
    unsafeAtomicAdd(dp + 2, wgt * v.z);
    unsafeAtomicAdd(dp + 3, wgt * v.w);
}

// x = lrelu(agg); xs += x
__global__ void lrelu_acc_kernel(float* __restrict__ xbuf, const float* __restrict__ agg,
                                 float* __restrict__ xs, long n4)
{
    long t = (long)blockIdx.x * blockDim.x + threadIdx.x;
    if (t >= n4) return;
    float4 a = ((const float4*)agg)[t];
    a.x = lrelu_f(a.x); a.y = lrelu_f(a.y); a.z = lrelu_f(a.z); a.w = lrelu_f(a.w);
    ((float4*)xbuf)[t] = a;
    float4 s = ((float4*)xs)[t];
    s.x += a.x; s.y += a.y; s.z += a.z; s.w += a.w;
    ((float4*)xs)[t] = s;
}

// out[j,k] = sum_m A[j*lda + colOff + m] * B[m*128 + k]   (tiny weight-fold GEMMs)
__global__ void combo_kernel(const float* __restrict__ A, int lda, int colOff,
                             const float* __restrict__ B, float* __restrict__ out)
{
    int k = threadIdx.x;   // 128
    int j = blockIdx.x;
    float s = 0.f;
    for (int m = 0; m < 128; ++m) s += A[(size_t)j * lda + colOff + m] * B[m * 128 + k];
    out[(size_t)j * 128 + k] = s;
}

// batch is sorted: register-accumulate runs, flush with atomics on group change.
__global__ void segsum_kernel(const float* __restrict__ xc, const int* __restrict__ batch,
                              const float* __restrict__ states,
                              float* gsel, float* gsum, int nrows)
{
    int f = threadIdx.x;   // 0..127
    int start = blockIdx.x * 64;
    if (start >= nrows) return;
    int end = start + 64; if (end > nrows) end = nrows;
    float asel = 0.f, asum = 0.f;
    int curg = batch[start];
    for (int i = start; i < end; ++i) {
        int g = batch[i];
        if (g != curg) {
            unsafeAtomicAdd(&gsum[curg * 128 + f], asum);
            unsafeAtomicAdd(&gsel[curg * 128 + f], asel);
            asum = 0.f; asel = 0.f; curg = g;
        }
        float v = xc[(size_t)i * 128 + f];
        asum += v;
        if (states[i] == 1.f) asel += v;
    }
    unsafeAtomicAdd(&gsum[curg * 128 + f], asum);
    unsafeAtomicAdd(&gsel[curg * 128 + f], asel);
}

// w[g,j] = gsel[g]·D1[j] + gsum[g]·D2[j]   (16x192 table)
__global__ void wcomb_kernel(const float* __restrict__ gsel, const float* __restrict__ gsum,
                             const float* __restrict__ D1, const float* __restrict__ D2,
                             float* __restrict__ w)
{
    int t = blockIdx.x * blockDim.x + threadIdx.x;
    if (t >= NG * 192) return;
    int g = t / 192, j = t % 192;
    float s = 0.f;
    for (int k = 0; k < 128; ++k)
        s += gsel[g * 128 + k] * D1[j * 128 + k] + gsum[g * 128 + k] * D2[j * 128 + k];
    w[t] = s;
}

// out[i] = hh[i,:192] · g4 + g4b   (one wave per node, shfl_xor wave32 reduction)
__global__ void reduce192_kernel(const float* __restrict__ hh, const float* __restrict__ g4,
                                 const float* __restrict__ g4b, float* __restrict__ out, int nrows)
{
    int node = blockIdx.x * 8 + (threadIdx.x >> 5);
    int lane = threadIdx.x & 31;
    if (node >= nrows) return;
    float s = 0.f;
    #pragma unroll
    for (int q = 0; q < 6; ++q) {
        int j = lane + 32 * q;
        s += hh[(size_t)node * 192 + j] * g4[j];
    }
    #pragma unroll
    for (int off = 16; off > 0; off >>= 1) s += __shfl_xor(s, off, 32);
    if (lane == 0) out[node] = s + g4b[0];
}

// ---------------------------------------------------------------------------
extern "C" void kernel_launch(void* const* d_in, const int* in_sizes, int n_in,
                              void* d_out, int out_size, void* d_ws, size_t ws_size,
                              hipStream_t stream)
{
    const float* x      = (const float*)d_in[0];
    const int*   eidx   = (const int*)  d_in[1];
    const float* ew     = (const float*)d_in[2];
    const int*   batch  = (const int*)  d_in[3];
    const float* states = (const float*)d_in[4];
    const float* pa0[2]  = { (const float*)d_in[5],  (const float*)d_in[10] };
    const float* pa1W[2] = { (const float*)d_in[6],  (const float*)d_in[11] };
    const float* pa1b[2] = { (const float*)d_in[7],  (const float*)d_in[12] };
    const float* plW[2]  = { (const float*)d_in[8],  (const float*)d_in[13] };
    const float* plb[2]  = { (const float*)d_in[9],  (const float*)d_in[14] };
    const float* beta0  = (const float*)d_in[15];
    const float* beta1  = (const float*)d_in[16];
    const float* beta2  = (const float*)d_in[17];
    const float* beta2b = (const float*)d_in[18];
    const float* g0     = (const float*)d_in[19];
    const float* g1     = (const float*)d_in[20];
    const float* g2     = (const float*)d_in[21];
    const float* g3     = (const float*)d_in[22];
    const float* g4     = (const float*)d_in[23];
    const float* g4b    = (const float*)d_in[24];

    float* ws = (float*)d_ws;
    const size_t NF = (size_t)NN * 128;
    float* x1   = ws;
    float* x2   = ws + 1 * NF;
    float* x1s  = ws + 2 * NF;
    float* x2s  = ws + 3 * NF;
    float* hbuf = ws + 4 * NF;   // h (T-loop) then xc (head)
    float* agg  = ws + 5 * NF;
    float* sm   = ws + 6 * NF;
    float* C1   = sm; sm += 128 * 128;
    float* C2   = sm; sm += 128 * 128;
    float* M0   = sm; sm += 192 * 128;
    float* D1   = sm; sm += 192 * 128;
    float* D2   = sm; sm += 192 * 128;
    float* gsel = sm; sm += NG * 128;
    float* gsum = sm; sm += NG * 128;
    float* wtab = sm; sm += NG * 192;
    float* hh   = x1s;           // reuse dead x1s/x2s region (N*192 <= 2*N*128)

    const int* s0 = eidx;        // edge_index[0]
    const int* s1 = eidx + EE;   // edge_index[1]
    float* xbufs[2] = { x1, x2 };
    float* xsums[2] = { x1s, x2s };

    const dim3 blk(256);
    const int gemmGrid = (NN + 127) / 128;
    const long n4 = (long)(NF / 4);
    const int zb = (int)((n4 + 255) / 256);
    const size_t sh128 = 128 * LDS_PITCH * sizeof(float);   // 66560 B
    const size_t sh192 = 192 * LDS_PITCH * sizeof(float);   // 99840 B

    split_kernel<<<(NN * 32 + 255) / 256, blk, 0, stream>>>(x, x1, x2, NN);
    zero4_kernel<<<zb, blk, 0, stream>>>((float4*)x1s, n4);
    zero4_kernel<<<zb, blk, 0, stream>>>((float4*)x2s, n4);

    for (int t = 0; t < 3; ++t) {
        for (int p = 0; p < 2; ++p) {
            const int* src = (p == 0) ? s1 : s0;
            const int* dst = (p == 0) ? s0 : s1;
            // h = lrelu((x + states*a0) @ a1W^T + a1b)
            gemm128_kernel<true, true, false, true><<<gemmGrid, blk, sh128, stream>>>(
                xbufs[p], pa1W[p] + (size_t)t * 128 * 128, pa1b[p] + t * 128,
                states, pa0[p] + t, hbuf, NN);
            // h = h @ linW^T + linb   (row-local: safe in-place)
            gemm128_kernel<false, false, false, true><<<gemmGrid, blk, sh128, stream>>>(
                hbuf, plW[p] + (size_t)t * 128 * 128, plb[p] + t * 128,
                nullptr, nullptr, hbuf, NN);
            zero4_kernel<<<zb, blk, 0, stream>>>((float4*)agg, n4);
            scatter_kernel<<<(EE * 32 + 255) / 256, blk, 0, stream>>>(hbuf, ew, src, dst, agg, EE);
            lrelu_acc_kernel<<<zb, blk, 0, stream>>>(xbufs[p], agg, xsums[p], n4);
        }
    }

    // Fold weight chains: C1=beta2[:, :128]@beta0, C2=beta2[:,128:]@beta1,
    // M0=g3[:, :128]@g0, D1=g3[:,128:256]@g1, D2=g3[:,256:]@g2
    combo_kernel<<<128, 128, 0, stream>>>(beta2, 256, 0,   beta0, C1);
    combo_kernel<<<128, 128, 0, stream>>>(beta2, 256, 128, beta1, C2);
    combo_kernel<<<192, 128, 0, stream>>>(g3, 384, 0,   g0, M0);
    combo_kernel<<<192, 128, 0, stream>>>(g3, 384, 128, g1, D1);
    combo_kernel<<<192, 128, 0, stream>>>(g3, 384, 256, g2, D2);

    // xc = lrelu(x1s@C1^T + x2s@C2^T + beta2_b)  -> hbuf
    gemm128_kernel<false, false, false, false><<<gemmGrid, blk, sh128, stream>>>(
        x1s, C1, nullptr, nullptr, nullptr, hbuf, NN);
    gemm128_kernel<false, true, true, true><<<gemmGrid, blk, sh128, stream>>>(
        x2s, C2, beta2b, nullptr, nullptr, hbuf, NN);

    zero4_kernel<<<(NG * 128 / 4 + 255) / 256, blk, 0, stream>>>((float4*)gsel, NG * 128 / 4);
    zero4_kernel<<<(NG * 128 / 4 + 255) / 256, blk, 0, stream>>>((float4*)gsum, NG * 128 / 4);
    segsum_kernel<<<(NN + 63) / 64, 128, 0, stream>>>(hbuf, batch, states, gsel, gsum, NN);
    wcomb_kernel<<<(NG * 192 + 255) / 256, blk, 0, stream>>>(gsel, gsum, D1, D2, wtab);

    // hh = lrelu(xc @ M0^T + wtab[batch]); out = hh . g4 + g4b
    gemm192_kernel<<<gemmGrid, blk, sh192, stream>>>(hbuf, M0, wtab, batch, hh, NN);
    reduce192_kernel<<<(NN + 7) / 8, blk, 0, stream>>>(hh, g4, g4b, (float*)d_out, NN);
}